// SequenceWindows_57037165691079
// MI455X (gfx1250) — compile-verified
//
#include <hip/hip_runtime.h>

// ---------------------------------------------------------------------------
// SequenceWindows: x(16,3,64,10000) f32 -> windows (736,3,64,1000) + labels(736)
// WINDOW_SIZE=1000, WINDOW_STRIDE=200, num_new=46.
// Pure bandwidth problem: 565 MB out + 123 MB compulsory in (input fits in
// 192MB L2, window overlap re-reads hit L2). Data path: gfx1250 async
// global->LDS->global b128 copies (ASYNCcnt), no VGPR staging.
// ---------------------------------------------------------------------------

#define NUM_NEW      46u
#define ROW_F4       250u       // 1000 floats / 4 per output row
#define SRC_ROW_F4   2500u      // 10000 floats / 4 per input row
#define STRIDE_F4    50u        // 200 floats / 4 window stride
#define F4_PER_BLOCK 1024u      // 256 threads x 4 float4
#define TOTAL_F4     35328000u  // 141312 rows * 250 f4
#define N_BLOCKS     34500      // TOTAL_F4 / F4_PER_BLOCK (exact)
#define SAMPLES_ELEMS 141312000
#define LABELS_N     736

// The gfx1250 async builtins take "int __vector_size__(16) *" operands
// (per hipcc diagnostic), in global (AS1) / LDS (AS3) address spaces.
typedef int v4i __attribute__((ext_vector_type(4)));
typedef __attribute__((address_space(1))) v4i as1_v4i;
typedef __attribute__((address_space(3))) v4i as3_v4i;

#if defined(__AMDGCN__) &&                                            \
    __has_builtin(__builtin_amdgcn_global_load_async_to_lds_b128) &&  \
    __has_builtin(__builtin_amdgcn_global_store_async_from_lds_b128)
#define USE_ASYNC_BUILTINS 1
#else
#define USE_ASYNC_BUILTINS 0
#endif

__device__ __forceinline__ void async_g2l_b128(const void* g, void* l) {
#if USE_ASYNC_BUILTINS
  __builtin_amdgcn_global_load_async_to_lds_b128(
      (as1_v4i*)(unsigned long long)g,
      (as3_v4i*)(unsigned)(unsigned long long)l, 0, 0);
#else
  unsigned lds_off = (unsigned)(unsigned long long)l;  // addr[31:0] == LDS offset
  asm volatile("global_load_async_to_lds_b128 %0, %1, off"
               :: "v"(lds_off), "v"(g) : "memory");
#endif
}

__device__ __forceinline__ void async_l2g_b128(void* g, const void* l) {
#if USE_ASYNC_BUILTINS
  __builtin_amdgcn_global_store_async_from_lds_b128(
      (as1_v4i*)(unsigned long long)g,
      (as3_v4i*)(unsigned)(unsigned long long)l, 0, 0);
#else
  unsigned lds_off = (unsigned)(unsigned long long)l;
  asm volatile("global_store_async_from_lds_b128 %0, %1, off"
               :: "v"(g), "v"(lds_off) : "memory");
#endif
}

__device__ __forceinline__ void wait_async0() {
#if defined(__AMDGCN__) && __has_builtin(__builtin_amdgcn_s_wait_asynccnt)
  __builtin_amdgcn_s_wait_asynccnt(0);
#else
  asm volatile("s_wait_asynccnt 0" ::: "memory");
#endif
}

// One float4 per (lane, k); block covers 1024 consecutive output float4s.
// Output f4 index g -> (row = g/250, t4 = g%250); row -> (n, w, c, h);
// source f4 index = ((n*3+c)*64+h)*2500 + w*50 + t4 (all 16B aligned).
__global__ __launch_bounds__(256) void
SequenceWindows_async_copy_kernel(const float* __restrict__ x,
                                  float* __restrict__ out) {
  __shared__ float4 stage[F4_PER_BLOCK];  // 16 KB of the 320 KB WGP LDS
  const unsigned tid  = threadIdx.x;
  const unsigned base = blockIdx.x * F4_PER_BLOCK + tid;
  const float4* __restrict__ src4 = (const float4*)x;
  float4* __restrict__ dst4 = (float4*)out;

  unsigned srcIdx[4];
#pragma unroll
  for (int k = 0; k < 4; ++k) {
    const unsigned g   = base + (unsigned)k * 256u;
    const unsigned row = g / ROW_F4;
    const unsigned t4  = g - row * ROW_F4;
    const unsigned h   = row & 63u;
    const unsigned r1  = row >> 6;
    const unsigned c   = r1 % 3u;
    const unsigned r2  = r1 / 3u;
    const unsigned w   = r2 % NUM_NEW;
    const unsigned n   = r2 / NUM_NEW;
    srcIdx[k] = ((n * 3u + c) * 64u + h) * SRC_ROW_F4 + w * STRIDE_F4 + t4;
  }

  // Phase 1: async global -> LDS (each lane owns its 4 private 16B slots)
#pragma unroll
  for (int k = 0; k < 4; ++k)
    async_g2l_b128(src4 + srcIdx[k], &stage[(unsigned)k * 256u + tid]);

  wait_async0();  // LDS writes of the async loads complete

  // Phase 2: async LDS -> global (contiguous, coalesced b128 stores)
#pragma unroll
  for (int k = 0; k < 4; ++k)
    async_l2g_b128(dst4 + (base + (unsigned)k * 256u),
                   &stage[(unsigned)k * 256u + tid]);

  wait_async0();  // drain before LDS dealloc at wave end
}

__global__ __launch_bounds__(256) void
SequenceWindows_labels_kernel(const float* __restrict__ labels,
                              float* __restrict__ out_labels) {
  const int i = blockIdx.x * blockDim.x + threadIdx.x;
  if (i < LABELS_N) out_labels[i] = labels[i / (int)NUM_NEW];
}

extern "C" void kernel_launch(void* const* d_in, const int* in_sizes, int n_in,
                              void* d_out, int out_size, void* d_ws, size_t ws_size,
                              hipStream_t stream) {
  (void)in_sizes; (void)n_in; (void)out_size; (void)d_ws; (void)ws_size;
  const float* x      = (const float*)d_in[0];
  const float* labels = (const float*)d_in[1];
  float* out          = (float*)d_out;

  SequenceWindows_async_copy_kernel<<<N_BLOCKS, 256, 0, stream>>>(x, out);
  SequenceWindows_labels_kernel<<<(LABELS_N + 255) / 256, 256, 0, stream>>>(
      labels, out + SAMPLES_ELEMS);
}